// model_72911364817543
// MI455X (gfx1250) — compile-verified
//
#include <hip/hip_runtime.h>

#define N_NODES 10000
#define N_EDGES 160000
#define BATCH   64
#define ITERS   30
#define LEAK    0.01f

// ---------------------------------------------------------------------------
// CDNA5 async global->LDS copy (ASYNCcnt path) + wait helper.
// Low 32 bits of a generic pointer to __shared__ data == LDS byte offset.
// ---------------------------------------------------------------------------
__device__ __forceinline__ void async_load_f32_to_lds(const float* gptr, float* lptr)
{
    unsigned int laddr = (unsigned int)(unsigned long long)lptr;
    asm volatile("global_load_async_to_lds_b32 %0, %1, off"
                 :: "v"(laddr), "v"((unsigned long long)gptr)
                 : "memory");
}
__device__ __forceinline__ void wait_asynccnt0()
{
    asm volatile("s_wait_asynccnt 0" ::: "memory");
}

// ============================ CSR construction =============================

__global__ void zero_counts_kernel(int* __restrict__ counts)
{
    int i = blockIdx.x * blockDim.x + threadIdx.x;
    if (i < N_NODES) counts[i] = 0;
}

__global__ void hist_kernel(const int* __restrict__ dst, int* __restrict__ counts)
{
    int e = blockIdx.x * blockDim.x + threadIdx.x;
    if (e < N_EDGES) atomicAdd(&counts[dst[e]], 1);
}

// single-block exclusive prefix sum over 10000 counts -> rowStart, cursor
__global__ void scan_kernel(const int* __restrict__ counts,
                            int* __restrict__ rowStart,
                            int* __restrict__ cursor)
{
    __shared__ int sdata[1024];
    const int t = threadIdx.x;
    const int ITEMS = 10;                       // 1024*10 >= 10000
    int total = 0;
#pragma unroll
    for (int j = 0; j < ITEMS; ++j) {
        int i = t * ITEMS + j;
        total += (i < N_NODES) ? counts[i] : 0;
    }
    sdata[t] = total;
    __syncthreads();
    for (int off = 1; off < 1024; off <<= 1) {  // Hillis-Steele inclusive scan
        int v = (t >= off) ? sdata[t - off] : 0;
        __syncthreads();
        sdata[t] += v;
        __syncthreads();
    }
    int running = (t == 0) ? 0 : sdata[t - 1];
#pragma unroll
    for (int j = 0; j < ITEMS; ++j) {
        int i = t * ITEMS + j;
        if (i < N_NODES) {
            rowStart[i] = running;
            cursor[i]   = running;
            running    += counts[i];
        }
    }
    if (t == 1023) rowStart[N_NODES] = sdata[1023];   // == N_EDGES
}

__global__ void fill_kernel(const int* __restrict__ dst, const int* __restrict__ src,
                            const float* __restrict__ weights,
                            int* __restrict__ cursor,
                            int* __restrict__ eid, int* __restrict__ colI,
                            float* __restrict__ wE)
{
    int e = blockIdx.x * blockDim.x + threadIdx.x;
    if (e >= N_EDGES) return;
    int ep = (e + 2048 < N_EDGES) ? e + 2048 : e;
    __builtin_prefetch(dst + ep, 0, 1);
    __builtin_prefetch(src + ep, 0, 1);
    int d    = dst[e];
    int slot = atomicAdd(&cursor[d], 1);
    eid[slot]  = e;
    colI[slot] = src[e];
    wE[slot]   = weights[e];
}

// canonicalize per-row edge order (sort by original edge id) -> bitwise
// deterministic accumulation order on every call. Rows are ~16 long.
__global__ void sort_kernel(const int* __restrict__ rowStart,
                            int* __restrict__ eid, int* __restrict__ colI,
                            float* __restrict__ wE)
{
    int i = blockIdx.x * blockDim.x + threadIdx.x;
    if (i >= N_NODES) return;
    int s0 = rowStart[i], s1 = rowStart[i + 1];
    for (int a = s0 + 1; a < s1; ++a) {         // insertion sort
        int   ke = eid[a];
        int   kc = colI[a];
        float kw = wE[a];
        int b = a - 1;
        while (b >= s0 && eid[b] > ke) {
            eid[b + 1]  = eid[b];
            colI[b + 1] = colI[b];
            wE[b + 1]   = wE[b];
            --b;
        }
        eid[b + 1] = ke; colI[b + 1] = kc; wE[b + 1] = kw;
    }
}

// ============================ compute kernels ==============================

// bIn = x^T + bias via async-to-LDS tiled transpose; xhat = leaky(bIn)
// (this IS iteration 1 since xhat0 == 0). Tile 32 nodes x 64 batch, pitch 65.
__global__ void init_kernel(const float* __restrict__ x,
                            const float* __restrict__ bias,
                            float* __restrict__ bIn,
                            float* __restrict__ xhat)
{
    __shared__ float tile[32 * 65];
    const int t  = threadIdx.x;
    const int n0 = blockIdx.x * 32;

#pragma unroll
    for (int k = 0; k < 8; ++k) {               // 2048 elems / 256 threads
        int idx = k * 256 + t;
        int ln  = idx & 31;                     // node fast -> coalesced x read
        int b   = idx >> 5;
        int n   = n0 + ln;
        if (n < N_NODES)
            async_load_f32_to_lds(&x[b * N_NODES + n], &tile[ln * 65 + b]);
    }
    wait_asynccnt0();
    __syncthreads();
#pragma unroll
    for (int k = 0; k < 8; ++k) {
        int idx = k * 256 + t;
        int b   = idx & 63;                     // batch fast -> coalesced write
        int ln  = idx >> 6;
        int n   = n0 + ln;
        if (n < N_NODES) {
            float v = tile[ln * 65 + b] + bias[n];
            int o   = n * BATCH + b;
            bIn[o]  = v;
            xhat[o] = (v < 0.0f) ? LEAK * v : v;
        }
    }
}

// fused CSR SpMM + bias + leakyReLU: one wave32 per dst node, zero atomics.
// Each lane owns a float2 of the 64-wide batch row. Edge metadata is loaded
// cooperatively in chunks of 32, broadcast via __shfl, and the gather is
// software-pipelined (row j+1 in flight while FMA-ing row j).
__global__ void spmm_kernel(const float* __restrict__ xcur,
                            float* __restrict__ xnext,
                            const float* __restrict__ bIn,
                            const int* __restrict__ rowStart,
                            const int* __restrict__ colI,
                            const float* __restrict__ wE)
{
    int wid = (blockIdx.x * blockDim.x + threadIdx.x) >> 5;
    wid = __builtin_amdgcn_readfirstlane(wid);      // wave-uniform -> SGPR
    const int lane = threadIdx.x & 31;
    if (wid >= N_NODES) return;

    const int s0 = rowStart[wid], s1 = rowStart[wid + 1];

    // hoist bias row load: latency overlaps the whole gather loop
    const float2 bv = ((const float2*)(bIn + wid * BATCH))[lane];

    float accx = 0.0f, accy = 0.0f;

    for (int base = s0; base < s1; base += 32) {
        int   k  = base + lane;
        float wv = 0.0f;
        int   sv = 0;
        if (k < s1) { wv = wE[k]; sv = colI[k]; }
        const int cnt = min(32, s1 - base);

        // pipelined gather: prologue loads row 0
        float  wj = __shfl(wv, 0, 32);
        int    sj = __shfl(sv, 0, 32);
        float2 xv = ((const float2*)(xcur + sj * BATCH))[lane];

        for (int j = 0; j < cnt; ++j) {
            const float  wc = wj;
            const float2 xc = xv;
            if (j + 1 < cnt) {                      // issue next row's load
                wj = __shfl(wv, j + 1, 32);
                sj = __shfl(sv, j + 1, 32);
                xv = ((const float2*)(xcur + sj * BATCH))[lane];
            }
            accx = fmaf(wc, xc.x, accx);
            accy = fmaf(wc, xc.y, accy);
        }
    }

    float rx = accx + bv.x;
    float ry = accy + bv.y;
    rx = (rx < 0.0f) ? LEAK * rx : rx;
    ry = (ry < 0.0f) ? LEAK * ry : ry;
    float2 r; r.x = rx; r.y = ry;
    ((float2*)(xnext + wid * BATCH))[lane] = r;
}

// out[b*N + n] = xhat[n*B + b] via async-to-LDS tiled transpose
__global__ void out_kernel(const float* __restrict__ xhat,
                           float* __restrict__ out)
{
    __shared__ float tile[32 * 65];
    const int t  = threadIdx.x;
    const int n0 = blockIdx.x * 32;

#pragma unroll
    for (int k = 0; k < 8; ++k) {
        int idx = k * 256 + t;
        int b   = idx & 63;                     // batch fast -> coalesced read
        int ln  = idx >> 6;
        int n   = n0 + ln;
        if (n < N_NODES)
            async_load_f32_to_lds(&xhat[n * BATCH + b], &tile[ln * 65 + b]);
    }
    wait_asynccnt0();
    __syncthreads();
#pragma unroll
    for (int k = 0; k < 8; ++k) {
        int idx = k * 256 + t;
        int ln  = idx & 31;                     // node fast -> coalesced write
        int b   = idx >> 5;
        int n   = n0 + ln;
        if (n < N_NODES) out[b * N_NODES + n] = tile[ln * 65 + b];
    }
}

// ---------------------------------------------------------------------------
extern "C" void kernel_launch(void* const* d_in, const int* in_sizes, int n_in,
                              void* d_out, int out_size, void* d_ws, size_t ws_size,
                              hipStream_t stream)
{
    const float* x       = (const float*)d_in[0];   // [64, 10000]
    const float* weights = (const float*)d_in[1];   // [160000]
    const float* bias    = (const float*)d_in[2];   // [10000]
    const int*   dst     = (const int*)d_in[3];     // [160000]
    const int*   src     = (const int*)d_in[4];     // [160000]
    float*       out     = (float*)d_out;           // [64, 10000]

    float* bIn      = (float*)d_ws;                 // 640000 f32
    float* xhatA    = bIn   + N_NODES * BATCH;      // 640000 f32
    float* xhatB    = xhatA + N_NODES * BATCH;      // 640000 f32
    float* wE       = xhatB + N_NODES * BATCH;      // 160000 f32
    int*   colI     = (int*)(wE + N_EDGES);         // 160000 i32
    int*   eid      = colI + N_EDGES;               // 160000 i32
    int*   rowStart = eid + N_EDGES;                // 10001 i32
    int*   cursor   = rowStart + (N_NODES + 1);     // 10000 i32
    int*   counts   = cursor + N_NODES;             // 10000 i32

    const int nodeBlocks = (N_NODES + 255) / 256;   // 40
    const int edgeBlocks = (N_EDGES + 255) / 256;   // 625
    const int tiles      = (N_NODES + 31) / 32;     // 313
    const int spmmBlocks = (N_NODES + 7) / 8;       // 1250 (8 waves/block)

    // ---- build CSR (deterministic structure, canonical per-row order) ----
    zero_counts_kernel<<<nodeBlocks, 256, 0, stream>>>(counts);
    hist_kernel<<<edgeBlocks, 256, 0, stream>>>(dst, counts);
    scan_kernel<<<1, 1024, 0, stream>>>(counts, rowStart, cursor);
    fill_kernel<<<edgeBlocks, 256, 0, stream>>>(dst, src, weights, cursor,
                                                eid, colI, wE);
    sort_kernel<<<nodeBlocks, 256, 0, stream>>>(rowStart, eid, colI, wE);

    // ---- iteration 1 fused into init ----
    init_kernel<<<tiles, 256, 0, stream>>>(x, bias, bIn, xhatA);

    // ---- iterations 2..30: fused gather-SpMM + bias + leakyReLU ----
    const float* cur = xhatA;
    float*       nxt = xhatB;
    for (int it = 1; it < ITERS; ++it) {
        spmm_kernel<<<spmmBlocks, 256, 0, stream>>>(cur, nxt, bIn,
                                                    rowStart, colI, wE);
        float* tmp = (float*)cur; cur = nxt; nxt = tmp;
    }

    out_kernel<<<tiles, 256, 0, stream>>>(cur, out);
}